// SpikeFP16Adder_29875792511519
// MI455X (gfx1250) — compile-verified
//
#include <hip/hip_runtime.h>
#include <hip/hip_fp16.h>
#include <stdint.h>

// ---------------------------------------------------------------------------
// SpikeFP16Adder for MI455X (gfx1250).
// Memory-bound streaming op (768 MiB/call -> ~34us floor @ 23.3 TB/s).
// Uses the CDNA5 async global<->LDS path (ASYNCcnt) for fully coalesced
// 512B-per-wave-instruction transfers + LDS transpose of the 64B rows.
// ---------------------------------------------------------------------------

#define THREADS        256          // 8 wave32 per block
#define ROWS_PER_BLOCK 256          // one row per thread
#define ROW_DW         20           // 16 data dwords + 4 pad dwords (bank decorrelation)
#define ROW_BYTES      (ROW_DW * 4) // 80 B per staged row

// --- CDNA5 async-tensor path (inline asm: portable across ROCm7.2/clang-23) ---
__device__ __forceinline__ void async_load_b128(uint32_t lds_addr, uint32_t goff,
                                                uint64_t sbase) {
  // VDST = LDS byte address, VADDR = per-lane byte offset, SADDR = scalar base
  asm volatile("global_load_async_to_lds_b128 %0, %1, %2"
               :: "v"(lds_addr), "v"(goff), "s"(sbase) : "memory");
}
__device__ __forceinline__ void async_store_b128(uint32_t goff, uint32_t lds_addr,
                                                 uint64_t sbase) {
  asm volatile("global_store_async_from_lds_b128 %0, %1, %2"
               :: "v"(goff), "v"(lds_addr), "s"(sbase) : "memory");
}
__device__ __forceinline__ void wait_async0() {
  asm volatile("s_wait_asynccnt 0x0" ::: "memory");
}

// bit-exact fp16 add: fp32 add is exact for two fp16 operands (2p+2 <= 24),
// then a single v_cvt_f16_f32 (RNE, denorms allowed by default mode) rounds once.
__device__ __forceinline__ unsigned short fp16_add_bits(uint32_t ua, uint32_t ub) {
  float fa = __half2float(__ushort_as_half((unsigned short)ua));
  float fb = __half2float(__ushort_as_half((unsigned short)ub));
  return __half_as_ushort(__float2half(fa + fb));
}

__global__ __launch_bounds__(THREADS)
void spike_fp16_add_kernel(const float* __restrict__ A,
                           const float* __restrict__ B,
                           float* __restrict__ out) {
  __shared__ __align__(16) uint32_t ldsA[ROWS_PER_BLOCK * ROW_DW]; // 20 KB
  __shared__ __align__(16) uint32_t ldsB[ROWS_PER_BLOCK * ROW_DW]; // 20 KB

  const uint32_t t = threadIdx.x;
  const uint64_t blockByteBase = (uint64_t)blockIdx.x * (ROWS_PER_BLOCK * 64ull);

  const uint64_t aBase = (uint64_t)(uintptr_t)A + blockByteBase;
  const uint64_t bBase = (uint64_t)(uintptr_t)B + blockByteBase;
  const uint64_t oBase = (uint64_t)(uintptr_t)out + blockByteBase;

  // AS(3) byte offset of the staging buffers (generic shared ptr low 32 bits).
  const uint32_t ldsA0 = (uint32_t)(uintptr_t)(&ldsA[0]);
  const uint32_t ldsB0 = (uint32_t)(uintptr_t)(&ldsB[0]);

  // ---- Stage 1: coalesced async copy global -> LDS (tracked by ASYNCcnt) ----
  // chunk c (0..1023) = 16B of the 16 KB tile; row = c>>2, chunk-in-row = c&3.
#pragma unroll
  for (uint32_t k = 0; k < 4; ++k) {
    const uint32_t c    = t + k * THREADS;                       // consecutive per lane
    const uint32_t goff = c * 16u;                               // contiguous global
    const uint32_t loff = (c >> 2) * ROW_BYTES + (c & 3u) * 16u; // padded row layout
    async_load_b128(ldsA0 + loff, goff, aBase);
    async_load_b128(ldsB0 + loff, goff, bBase);
  }
  wait_async0();       // this wave's async writes landed in LDS
  __syncthreads();     // cross-wave visibility (rows span waves' chunks)

  // ---- Stage 2: per-thread row: pack -> fp16 add -> unpack --------------------
  const uint32_t* rowA = &ldsA[t * ROW_DW];
  const uint32_t* rowB = &ldsB[t * ROW_DW];
  uint32_t ua = 0u, ub = 0u;
#pragma unroll
  for (int j = 0; j < 16; ++j) {
    // bits are exactly 0.0f (0x00000000) or 1.0f (0x3F800000): bit23 is the flag
    ua |= ((rowA[j] >> 23) & 1u) << (15 - j);   // MSB-first [S,E4..E0,M9..M0]
    ub |= ((rowB[j] >> 23) & 1u) << (15 - j);
  }
  const unsigned short us = fp16_add_bits(ua, ub);

  // Reuse the A staging row (only thread t ever touches row t between barriers).
  uint32_t* rowO = &ldsA[t * ROW_DW];
#pragma unroll
  for (int j = 0; j < 16; ++j)
    rowO[j] = ((us >> (15 - j)) & 1u) ? 0x3F800000u : 0u;  // 1.0f / 0.0f patterns
  __syncthreads();     // includes dscnt drain: LDS results visible to all waves

  // ---- Stage 3: coalesced async store LDS -> global ---------------------------
#pragma unroll
  for (uint32_t k = 0; k < 4; ++k) {
    const uint32_t c    = t + k * THREADS;
    const uint32_t goff = c * 16u;
    const uint32_t loff = (c >> 2) * ROW_BYTES + (c & 3u) * 16u;
    async_store_b128(goff, ldsA0 + loff, oBase);
  }
  wait_async0();       // drain before wave retires (S_ENDPGM would also wait-idle)
}

// Scalar tail for row counts not divisible by ROWS_PER_BLOCK (unused for N=4194304).
__global__ void spike_fp16_add_tail(const float* __restrict__ A,
                                    const float* __restrict__ B,
                                    float* __restrict__ out,
                                    long long firstRow, long long totalRows) {
  long long row = firstRow + (long long)blockIdx.x * blockDim.x + threadIdx.x;
  if (row >= totalRows) return;
  const float* a = A + row * 16;
  const float* b = B + row * 16;
  uint32_t ua = 0u, ub = 0u;
#pragma unroll
  for (int j = 0; j < 16; ++j) {
    ua |= ((__float_as_uint(a[j]) >> 23) & 1u) << (15 - j);
    ub |= ((__float_as_uint(b[j]) >> 23) & 1u) << (15 - j);
  }
  const unsigned short us = fp16_add_bits(ua, ub);
  float* o = out + row * 16;
#pragma unroll
  for (int j = 0; j < 16; ++j)
    o[j] = __uint_as_float(((us >> (15 - j)) & 1u) ? 0x3F800000u : 0u);
}

extern "C" void kernel_launch(void* const* d_in, const int* in_sizes, int n_in,
                              void* d_out, int out_size, void* d_ws, size_t ws_size,
                              hipStream_t stream) {
  const float* A   = (const float*)d_in[0];
  const float* B   = (const float*)d_in[1];
  float*       out = (float*)d_out;

  const long long rows    = (long long)in_sizes[0] / 16;   // N = 4,194,304
  const long long nblocks = rows / ROWS_PER_BLOCK;         // 16384 for N
  if (nblocks > 0) {
    spike_fp16_add_kernel<<<(uint32_t)nblocks, THREADS, 0, stream>>>(A, B, out);
  }
  const long long rem = rows - nblocks * ROWS_PER_BLOCK;
  if (rem > 0) {
    const long long first = nblocks * ROWS_PER_BLOCK;
    spike_fp16_add_tail<<<(uint32_t)((rem + 255) / 256), 256, 0, stream>>>(
        A, B, out, first, rows);
  }
}